// SSESparseSoftmax_77154792505537
// MI455X (gfx1250) — compile-verified
//
#include <hip/hip_runtime.h>
#include <stdint.h>

// Problem constants (match reference): B=4, S=2048, D=256, P=32, K=8
#define DDIM 256
#define PDIM 32
#define KIDX 8

__global__ __launch_bounds__(256) void sse_sparse_softmax_kernel(
    const float* __restrict__ keys,
    const int*   __restrict__ pidx,
    float*       __restrict__ out)
{
    // 32 KB staging tile, laid out exactly like the row's output block [p][d]
    __shared__ __align__(16) float tile[PDIM * DDIM];

    const int row = blockIdx.x;      // flattened (b*S + s), 0..8191
    const int d   = threadIdx.x;     // 0..255

    // ---- build partition mask (uniform per block -> scalar loads) ----
    const int* ip = pidx + row * KIDX;
    unsigned m = 0u;
#pragma unroll
    for (int i = 0; i < KIDX; ++i) m |= 1u << (ip[i] & 31);

    // ---- per-element softmax via single exp + geometric recurrence ----
    const float k   = keys[row * DDIM + d];
    const bool  pos = (k >= 0.0f);
    const float r   = __expf(-fabsf(k));        // ratio in (0, 1]
    // value at step j (counted from the max end) is r^j for both signs;
    // destination p is j (k<0) or 31-j (k>=0): remap mask with bit-reverse.
    const unsigned mt = pos ? __brev(m) : m;

    // pass 1: E = sum_j r^j (full softmax denom), Sm = masked sum
    float A = 1.0f, E = 0.0f, Sm = 0.0f;
#pragma unroll
    for (int j = 0; j < PDIM; ++j) {
        E  += A;
        Sm += ((mt >> j) & 1u) ? A : 0.0f;
        A  *= r;
    }
    const float eps = 1.1920929e-7f;            // FLT_EPSILON, as reference
    const float f   = 1.0f / (8.0f * (Sm + eps * E));  // includes /K

    // pass 2: stage outputs into LDS (bank-conflict-free: bank = d mod 64)
    int       base = pos ? ((PDIM - 1) * DDIM + d) : d;
    const int step = pos ? -DDIM : DDIM;
    float Af = f;                                // fold normalizer into A
#pragma unroll
    for (int j = 0; j < PDIM; ++j) {
        tile[base] = ((mt >> j) & 1u) ? Af : 0.0f;
        base += step;
        Af   *= r;
    }

    __syncthreads();

    // ---- async LDS -> global: 32 KB row tile, b128 per lane, coalesced,
    //      non-temporal (write-once 256 MB stream; don't thrash 192 MB L2) ----
    const uint64_t gbase = (uint64_t)(uintptr_t)out
                         + (uint64_t)row * (PDIM * DDIM * 4);
    const unsigned lbase = (unsigned)(uintptr_t)(&tile[0]);
#pragma unroll
    for (int i = 0; i < 8; ++i) {
        const unsigned c = (unsigned)(i * 256 + d) * 16u;   // chunk byte offset
        uint64_t gaddr = gbase + c;                         // per-lane global addr
        unsigned laddr = lbase + c;                         // per-lane LDS addr
        asm volatile("global_store_async_from_lds_b128 %0, %1, off th:TH_STORE_NT"
                     :: "v"(gaddr), "v"(laddr)
                     : "memory");
    }
    // Outstanding ASYNCcnt drains at s_endpgm's implicit wait-idle.
}

extern "C" void kernel_launch(void* const* d_in, const int* in_sizes, int n_in,
                              void* d_out, int out_size, void* d_ws, size_t ws_size,
                              hipStream_t stream) {
    const float* keys = (const float*)d_in[0];   // (4, 2048, 256) f32
    const int*   pidx = (const int*)d_in[1];     // (4, 2048, 8) i32
    float*       out  = (float*)d_out;           // (4, 2048, 32, 256) f32

    const int rows = in_sizes[0] / DDIM;         // B*S = 8192
    sse_sparse_softmax_kernel<<<rows, 256, 0, stream>>>(keys, pidx, out);
}